// Mamba2dBlock_83923660964613
// MI455X (gfx1250) — compile-verified
//
#include <hip/hip_runtime.h>
#include <hip/hip_bf16.h>
#include <math.h>

// Problem dims (from reference)
#define B_   2
#define L_   4096
#define DM   192      // d_model
#define DS   16       // d_state
#define DC   4        // d_conv
#define DI   384      // d_inner
#define DTR  12       // dt_rank
#define NSEQ 8        // 4 directions * B_
#define HW   64       // sqrt(L_)

// GEMM tiling
#define BM 128
#define BN 64
#define BK 32
#define LDA_P 36      // LDS stride (floats) for A tile: 16 rows -> 16 distinct 4-bank groups
#define LDB_P 72      // LDS stride (floats) for B tile: lane halves get disjoint bank windows

typedef __attribute__((ext_vector_type(2))) float v2f;
typedef __attribute__((ext_vector_type(8))) float v8f;
typedef __attribute__((ext_vector_type(4))) int   v4i;

#define AS1 __attribute__((address_space(1)))
#define AS3 __attribute__((address_space(3)))

#if __has_builtin(__builtin_amdgcn_global_load_async_to_lds_b128)
#define HAVE_ASYNC_LDS 1
#else
#define HAVE_ASYNC_LDS 0
#endif

__device__ __forceinline__ float sigmoidf_(float v) { return 1.0f / (1.0f + __expf(-v)); }

__device__ __forceinline__ void async_cp16(const float* g, float* l) {
#if HAVE_ASYNC_LDS
  // Builtin signature (from diagnostics): (v4i addrspace(1)*, v4i addrspace(3)*, imm offset, imm cpol)
  __builtin_amdgcn_global_load_async_to_lds_b128((AS1 v4i*)g, (AS3 v4i*)l, 0, 0);
#else
  *(float4*)l = *(const float4*)g;
#endif
}

__device__ __forceinline__ void wait_async_lds() {
#if HAVE_ASYNC_LDS
#if __has_builtin(__builtin_amdgcn_s_wait_asynccnt)
  __builtin_amdgcn_s_wait_asynccnt(0);
#else
  asm volatile("s_wait_asynccnt 0x0" ::: "memory");
#endif
#endif
}

// ---------------------------------------------------------------------------
// Build the 4-direction batched input X [8, L, DM] from x [2, L, DM]
// ---------------------------------------------------------------------------
__global__ void build_x_kernel(const float* __restrict__ x, float* __restrict__ X) {
  int idx = blockIdx.x * blockDim.x + threadIdx.x;
  const int total = NSEQ * L_ * DM;
  if (idx >= total) return;
  int d = idx % DM;
  int t = (idx / DM) % L_;
  int n = idx / (DM * L_);
  int b = n & 1, dir = n >> 1;
  int p = (dir & 2) ? (L_ - 1 - t) : t;           // dirs 2,3 = sequence flip
  int src;
  if (dir & 1) { int r = p >> 6, c = p & 63; src = (HW - 1 - c) * HW + r; }  // x1[b, 63-c, r]
  else         { src = p; }
  X[idx] = x[((size_t)b * L_ + src) * DM + d];
}

// ---------------------------------------------------------------------------
// Zero-pad x_proj_w [384,44] -> [384,64] and dt_proj_w [12,384] -> [32,384]
// (tail-free GEMMs: garbage A columns always multiply zero B rows)
// ---------------------------------------------------------------------------
__global__ void pad_weights_kernel(const float* __restrict__ xp, const float* __restrict__ dtw,
                                   float* __restrict__ xp_pad, float* __restrict__ dt_pad) {
  int idx = blockIdx.x * blockDim.x + threadIdx.x;
  const int NXP = DI * 64;
  if (idx < NXP) {
    int r = idx / 64, c = idx % 64;
    xp_pad[idx] = (c < (DTR + 2 * DS)) ? xp[r * (DTR + 2 * DS) + c] : 0.0f;
  }
  int j = idx - NXP;
  if (j >= 0 && j < 32 * DI) {
    int r = j / DI, c = j % DI;
    dt_pad[j] = (r < DTR) ? dtw[r * DI + c] : 0.0f;
  }
}

// ---------------------------------------------------------------------------
// f32 WMMA GEMM, LDS double-buffered, async global->LDS staging.
// C[M,N] = A[M,K] * B[K,N], requires M%128==0, N%64==0, K%32==0 (all callers).
// Block = 256 threads = 8 waves; wave w computes rows [m0+16w, m0+16w+16) x 64 cols.
// EPI: 0 none, 1 +bias,softplus, 2 +bias,+residual
// ---------------------------------------------------------------------------
template <int EPI>
__global__ __launch_bounds__(256) void wmma_gemm_kernel(
    const float* __restrict__ A, int lda,
    const float* __restrict__ Bw, int ldb,
    float* __restrict__ C, int ldc, int K,
    const float* __restrict__ bias,
    const float* __restrict__ resid, int ldr) {
  __shared__ float sA[2][BM * LDA_P];
  __shared__ float sB[2][BK * LDB_P];

  const int tid  = threadIdx.x;
  const int lane = tid & 31;
  const int wave = tid >> 5;
  const int half = lane >> 4;    // which K-pair this lane holds in a WMMA fragment
  const int l16  = lane & 15;
  const int m0 = blockIdx.y * BM;
  const int n0 = blockIdx.x * BN;

  const float* Abase = A + (size_t)m0 * lda;
  const float* Bbase = Bw + n0;

  // staging coordinates (float4 granularity, coalesced: f = q*256 + tid)
  int arow[4], ac4[4];
#pragma unroll
  for (int q = 0; q < 4; q++) { int f = q * 256 + tid; arow[q] = f >> 3; ac4[q] = f & 7; }
  int brow[2], bc4[2];
#pragma unroll
  for (int q = 0; q < 2; q++) { int f = q * 256 + tid; brow[q] = f >> 4; bc4[q] = f & 15; }

  auto stage = [&](int k0, int buf) {
#pragma unroll
    for (int q = 0; q < 4; q++)
      async_cp16(Abase + (size_t)arow[q] * lda + k0 + ac4[q] * 4,
                 &sA[buf][arow[q] * LDA_P + ac4[q] * 4]);
#pragma unroll
    for (int q = 0; q < 2; q++)
      async_cp16(Bbase + (size_t)(k0 + brow[q]) * ldb + bc4[q] * 4,
                 &sB[buf][brow[q] * LDB_P + bc4[q] * 4]);
  };

  v8f acc[4];
#pragma unroll
  for (int t = 0; t < 4; t++) acc[t] = (v8f){0.f, 0.f, 0.f, 0.f, 0.f, 0.f, 0.f, 0.f};

  const int KT = K / BK;
  stage(0, 0);
  wait_async_lds();
  __syncthreads();

  for (int kt = 0; kt < KT; kt++) {
    int cur = kt & 1;
    if (kt + 1 < KT) stage((kt + 1) * BK, cur ^ 1);

    const float* lA = &sA[cur][(wave * 16 + l16) * LDA_P + 2 * half];
    const float* lB = &sB[cur][0];
#pragma unroll
    for (int k = 0; k < BK; k += 4) {
      v2f a = *(const v2f*)(lA + k);                 // ds_load_b64, bank-conflict free
#pragma unroll
      for (int t = 0; t < 4; t++) {
        int col = t * 16 + l16;
        v2f b;
        b.x = lB[(k + 2 * half) * LDB_P + col];
        b.y = lB[(k + 2 * half + 1) * LDB_P + col];
        acc[t] = __builtin_amdgcn_wmma_f32_16x16x4_f32(
            false, a, false, b, (short)0, acc[t], false, false);
      }
    }
    wait_async_lds();
    __syncthreads();
  }

  const int mw = m0 + wave * 16;
#pragma unroll
  for (int t = 0; t < 4; t++) {
    int col = n0 + t * 16 + l16;
#pragma unroll
    for (int v = 0; v < 8; v++) {
      int row = mw + v + 8 * half;                  // C/D layout: VGPR v -> M=v (+8 hi lanes)
      float val = acc[t][v];
      if (EPI == 1) {                               // bias + softplus
        val += bias[col];
        val = (val > 20.0f) ? val : __logf(1.0f + __expf(val));
      } else if (EPI == 2) {                        // bias + residual
        val += bias[col] + resid[(size_t)row * ldr + col];
      }
      C[(size_t)row * ldc + col] = val;
    }
  }
}

// ---------------------------------------------------------------------------
// Causal depthwise conv (4 taps) + SiLU on the xc half of xz (cols 0..383 of 768)
// ---------------------------------------------------------------------------
__global__ void conv_silu_kernel(const float* __restrict__ xz, const float* __restrict__ w,
                                 const float* __restrict__ b, float* __restrict__ xcs) {
  int idx = blockIdx.x * blockDim.x + threadIdx.x;
  const int total = NSEQ * L_ * DI;
  if (idx >= total) return;
  int d = idx % DI;
  int t = (idx / DI) % L_;
  int n = idx / (DI * L_);
  const float* base = xz + (size_t)n * L_ * (2 * DI) + d;
  float acc = b[d];
#pragma unroll
  for (int k = 0; k < DC; k++) {
    int tt = t - (DC - 1) + k;
    if (tt >= 0) acc += base[(size_t)tt * (2 * DI)] * w[d * DC + k];
  }
  xcs[idx] = acc * sigmoidf_(acc);
}

// ---------------------------------------------------------------------------
// Selective scan. One lane per channel, 16 states in VGPRs, B/C staged in LDS
// in 128-step tiles. Fused epilogue: yact = (y + xc*D) * silu(z). In-place on xcs.
// ---------------------------------------------------------------------------
__global__ __launch_bounds__(128) void scan_kernel(
    const float* __restrict__ xz,    // dt in cols 0..383 (written by dt_proj), z in cols 384..767
    const float* __restrict__ xcs,   // conv+silu output
    const float* __restrict__ dbc,   // [N*L, 64]: B at cols 12..27, C at cols 28..43
    const float* __restrict__ A_log,
    const float* __restrict__ Dp,
    float* __restrict__ yact) {      // aliases xcs (read-before-write per element)
  __shared__ float bc[128 * 32];
  const int n  = blockIdx.x;
  const int ch = blockIdx.y * 128 + threadIdx.x;

  float a[DS], h[DS];
#pragma unroll
  for (int s = 0; s < DS; s++) { a[s] = -__expf(A_log[ch * DS + s]); h[s] = 0.0f; }
  const float dch = Dp[ch];
  const size_t row0 = (size_t)n * L_;

  for (int t0 = 0; t0 < L_; t0 += 128) {
    {  // cooperative stage of B,C (32 contiguous floats starting at col 12)
      int tr = threadIdx.x;
      const float4* src = (const float4*)(dbc + (row0 + t0 + tr) * 64 + 12);
      float4* dst = (float4*)(bc + tr * 32);
#pragma unroll
      for (int q = 0; q < 8; q++) dst[q] = src[q];
    }
    __syncthreads();
    for (int t = 0; t < 128; t++) {
      size_t r = row0 + t0 + t;
      float dtv = xz[r * (2 * DI) + ch];
      float xv  = xcs[r * DI + ch];
      float zv  = xz[r * (2 * DI) + DI + ch];
      float dtx = dtv * xv;
      const float* bcr = bc + t * 32;
      float y = 0.0f;
#pragma unroll
      for (int s = 0; s < DS; s++) {
        float dA = __expf(dtv * a[s]);      // a[s] = -exp(A_log) < 0
        h[s] = dA * h[s] + dtx * bcr[s];
        y += h[s] * bcr[16 + s];
      }
      yact[r * DI + ch] = (y + xv * dch) * (zv * sigmoidf_(zv));
    }
    __syncthreads();
  }
}

// ---------------------------------------------------------------------------
// Combine 4 directions (inverse permutations) + LayerNorm over DM=192.
// One wave32 per pixel, 6 channels per lane, __shfl_xor reduction.
// ---------------------------------------------------------------------------
__global__ __launch_bounds__(256) void combine_ln_kernel(
    const float* __restrict__ Y,     // [NSEQ*L, DM] out_proj result
    const float* __restrict__ g, const float* __restrict__ bta,
    float* __restrict__ ybar) {
  int wave = threadIdx.x >> 5, lane = threadIdx.x & 31;
  int P = blockIdx.x * 8 + wave;     // pixel 0..8191
  int b = P >> 12;                   // / 4096
  int p = P & 4095;
  int i = p >> 6, j = p & 63;
  int p2 = j * 64 + (HW - 1 - i);    // inverse of dir1 mapping
  size_t r1 = ((size_t)(0 + b) * L_ + p)          * DM;
  size_t r2 = ((size_t)(2 + b) * L_ + p2)         * DM;
  size_t r3 = ((size_t)(4 + b) * L_ + (L_-1-p))   * DM;
  size_t r4 = ((size_t)(6 + b) * L_ + (L_-1-p2))  * DM;

  float v[6];
  float s1 = 0.0f, s2 = 0.0f;
#pragma unroll
  for (int q = 0; q < 6; q++) {
    int d = lane + 32 * q;
    float vv = Y[r1 + d] + Y[r2 + d] + Y[r3 + d] + Y[r4 + d];
    v[q] = vv; s1 += vv; s2 += vv * vv;
  }
#pragma unroll
  for (int off = 16; off >= 1; off >>= 1) {
    s1 += __shfl_xor(s1, off, 32);
    s2 += __shfl_xor(s2, off, 32);
  }
  float mu   = s1 * (1.0f / DM);
  float var  = s2 * (1.0f / DM) - mu * mu;
  float rstd = rsqrtf(var + 1e-5f);
#pragma unroll
  for (int q = 0; q < 6; q++) {
    int d = lane + 32 * q;
    ybar[(size_t)P * DM + d] = (v[q] - mu) * rstd * g[d] + bta[d];
  }
}

// ---------------------------------------------------------------------------
extern "C" void kernel_launch(void* const* d_in, const int* in_sizes, int n_in,
                              void* d_out, int out_size, void* d_ws, size_t ws_size,
                              hipStream_t stream) {
  (void)in_sizes; (void)n_in; (void)out_size; (void)ws_size;
  const float* x         = (const float*)d_in[0];
  const float* in_proj_w = (const float*)d_in[1];   // [192, 768]
  const float* conv_w    = (const float*)d_in[2];   // [384, 4]
  const float* conv_b    = (const float*)d_in[3];
  const float* x_proj_w  = (const float*)d_in[4];   // [384, 44]
  const float* dt_proj_w = (const float*)d_in[5];   // [12, 384]
  const float* dt_proj_b = (const float*)d_in[6];
  const float* A_log     = (const float*)d_in[7];   // [384, 16]
  const float* Dp        = (const float*)d_in[8];
  const float* out_w     = (const float*)d_in[9];   // [384, 192]
  const float* ln_g      = (const float*)d_in[10];
  const float* ln_b      = (const float*)d_in[11];
  const float* blk_w     = (const float*)d_in[12];  // [192, 192]
  const float* blk_b     = (const float*)d_in[13];
  float* out = (float*)d_out;

  // Workspace layout (floats); aliasing keeps it ~191 MB.
  float* ws = (float*)d_ws;
  size_t off = 0;
  float* X      = ws + off; off += (size_t)NSEQ * L_ * DM;      // reused as Yout after out_proj
  float* xz     = ws + off; off += (size_t)NSEQ * L_ * 2 * DI;  // cols 0..383 later hold dt
  float* xcs    = ws + off; off += (size_t)NSEQ * L_ * DI;      // conv out, later yact (in-place)
  float* dbc    = ws + off; off += (size_t)NSEQ * L_ * 64;      // [.,64]: dt|B|C|zeros
  float* ybar   = ws + off; off += (size_t)B_ * L_ * DM;
  float* xp_pad = ws + off; off += (size_t)DI * 64;             // [384, 64]
  float* dt_pad = ws + off; off += (size_t)32 * DI;             // [32, 384]

  const int M = NSEQ * L_;  // 32768

  { int tot = NSEQ * L_ * DM;
    build_x_kernel<<<(tot + 255) / 256, 256, 0, stream>>>(x, X); }
  { int tot = DI * 64 + 32 * DI;
    pad_weights_kernel<<<(tot + 255) / 256, 256, 0, stream>>>(x_proj_w, dt_proj_w, xp_pad, dt_pad); }

  // xz[32768,768] = X[32768,192] @ in_proj_w[192,768]
  wmma_gemm_kernel<0><<<dim3((2 * DI) / BN, M / BM), 256, 0, stream>>>(
      X, DM, in_proj_w, 2 * DI, xz, 2 * DI, DM, nullptr, nullptr, 0);

  // causal depthwise conv + SiLU on xc half of xz
  { int tot = NSEQ * L_ * DI;
    conv_silu_kernel<<<(tot + 255) / 256, 256, 0, stream>>>(xz, conv_w, conv_b, xcs); }

  // dbc[32768,64] = xcs[32768,384] @ xp_pad[384,64]   (cols 44..63 come out zero)
  wmma_gemm_kernel<0><<<dim3(1, M / BM), 256, 0, stream>>>(
      xcs, DI, xp_pad, 64, dbc, 64, DI, nullptr, nullptr, 0);

  // dt = softplus(dbc[:, :32] @ dt_pad[32,384] + b); stored into xz cols 0..383
  wmma_gemm_kernel<1><<<dim3(DI / BN, M / BM), 256, 0, stream>>>(
      dbc, 64, dt_pad, DI, xz, 2 * DI, 32, dt_proj_b, nullptr, 0);

  // selective scan with fused (y + xc*D) * silu(z); in-place yact over xcs
  scan_kernel<<<dim3(NSEQ, DI / 128), 128, 0, stream>>>(xz, xcs, dbc, A_log, Dp, xcs);

  // Yout (reuse X) [32768,192] = yact[32768,384] @ out_w[384,192]
  wmma_gemm_kernel<0><<<dim3(DM / BN, M / BM), 256, 0, stream>>>(
      xcs, DI, out_w, DM, X, DM, DI, nullptr, nullptr, 0);

  // inverse permutations + sum 4 dirs + LayerNorm
  combine_ln_kernel<<<(B_ * L_) / 8, 256, 0, stream>>>(X, ln_g, ln_b, ybar);

  // out = x + ybar @ blk_w + blk_b
  wmma_gemm_kernel<2><<<dim3(DM / BN, (B_ * L_) / BM), 256, 0, stream>>>(
      ybar, DM, blk_w, DM, out, DM, DM, blk_b, x, DM);
}